// ComplexGAT_446676598989
// MI455X (gfx1250) — compile-verified
//
#include <hip/hip_runtime.h>
#include <hip/hip_bf16.h>

// ---------------------------------------------------------------------------
// ComplexGAT on MI455X (gfx1250): fp32 WMMA (16x16x4) for all GEMMs,
// branch-free WMMA inner loops (clamped A rows, templated skip),
// L2-resident edge gather/scatter with float atomics, fused epilogues.
// ---------------------------------------------------------------------------

typedef float v2f __attribute__((ext_vector_type(2)));
typedef float v8f __attribute__((ext_vector_type(8)));

#define C_DIM 128
#define G_DIM 512
#define NEG_SLOPE 0.2f
#define BN_EPS 1e-5f

// ---- helpers ---------------------------------------------------------------

__device__ __forceinline__ unsigned enc_f32(float f) {
    unsigned u = __float_as_uint(f);
    return (u & 0x80000000u) ? ~u : (u | 0x80000000u);
}
__device__ __forceinline__ float dec_f32(unsigned u) {
    return (u & 0x80000000u) ? __uint_as_float(u & 0x7fffffffu)
                             : __uint_as_float(~u);
}

__global__ void zero_f32_k(float* __restrict__ p, int n) {
    int i = blockIdx.x * 256 + threadIdx.x;
    if (i < n) p[i] = 0.0f;
}

// ---- 1. node GEMM: hl = (h [+ x0]) @ W   via v_wmma_f32_16x16x4_f32 --------
// block = 256 threads (8 waves). Each wave owns one 16-column tile of the
// 128-wide output; the block walks 8 row-tiles (128 rows of h).
// B (16 columns of W, 32 x float2 = 64 VGPRs) is hoisted out of the row-tile
// loop; A rows are clamped so the WMMA chain is completely branch-free.
template <int SKIP>
__global__ void gat_gemm_wmma(const float* __restrict__ h,
                              const float* __restrict__ x0,
                              const float* __restrict__ W,   // [128,128]
                              float* __restrict__ hl, int n_nodes) {
    const int wave = threadIdx.x >> 5;
    const int lane = threadIdx.x & 31;
    const int half = lane >> 4;     // lanes 16-31 hold K+2 / M+8 halves
    const int r    = lane & 15;
    const int row_block = blockIdx.x * 128;
    const int n0 = wave * 16;

    // hoist B: columns n0..n0+15 of W for all K, per the 4x16 fp32 layout
    v2f breg[32];
    {
        const float* bp = W + (size_t)(2 * half) * C_DIM + n0 + r;
#pragma unroll
        for (int t = 0; t < 32; ++t) {
            breg[t].x = bp[(size_t)(4 * t) * C_DIM];
            breg[t].y = bp[(size_t)(4 * t + 1) * C_DIM];
        }
    }

    for (int mt = 0; mt < 8; ++mt) {
        const int row0 = row_block + mt * 16;
        const int arow_real = row0 + r;
        const int arow = arow_real < n_nodes ? arow_real : n_nodes - 1; // clamp
        const float* ap = h + (size_t)arow * C_DIM + 2 * half;
        const float* xp = x0 + (size_t)arow * C_DIM + 2 * half;
        v8f acc = {};
#pragma unroll
        for (int t = 0; t < 32; ++t) {
            // A 16x4 fp32 layout: VGPR0 = K=2*half, VGPR1 = K=2*half+1
            v2f a;
            a.x = ap[4 * t];
            a.y = ap[4 * t + 1];
            if (SKIP) {
                a.x += xp[4 * t];
                a.y += xp[4 * t + 1];
            }
            acc = __builtin_amdgcn_wmma_f32_16x16x4_f32(
                false, a, false, breg[t], (short)0, acc, false, false);
        }
        // C/D layout: VGPR j -> M=j (lanes 0-15) / M=j+8 (lanes 16-31), N=r
#pragma unroll
        for (int j = 0; j < 8; ++j) {
            const int row = row0 + j + half * 8;
            if (row < n_nodes) hl[(size_t)row * C_DIM + n0 + r] = acc[j];
        }
    }
}

// ---- 2. per-node attention logits ------------------------------------------
__global__ void node_alpha_k(const float* __restrict__ hl,
                             const float* __restrict__ asrc,
                             const float* __restrict__ adst,
                             float* __restrict__ alpha_s,
                             float* __restrict__ alpha_d, int n) {
    const int node = (blockIdx.x * blockDim.x + threadIdx.x) >> 5;
    const int lane = threadIdx.x & 31;
    if (node >= n) return;
    const float* row = hl + (size_t)node * C_DIM;
    float s = 0.f, d = 0.f;
#pragma unroll
    for (int c = lane; c < C_DIM; c += 32) {
        const float v = row[c];
        s += v * asrc[c];
        d += v * adst[c];
    }
#pragma unroll
    for (int off = 16; off; off >>= 1) {
        s += __shfl_down(s, off, 32);
        d += __shfl_down(d, off, 32);
    }
    if (lane == 0) { alpha_s[node] = s; alpha_d[node] = d; }
}

// ---- 3a. segment max over destinations (monotone u32 encoding) -------------
__global__ void edge_max_k(const int* __restrict__ ei, int E, int ET,
                           const float* __restrict__ as,
                           const float* __restrict__ ad,
                           unsigned* __restrict__ emax_enc) {
    const int i = blockIdx.x * 256 + threadIdx.x;
    if (i >= ET) return;
    int s, d;
    if (i < E) { s = ei[i]; d = ei[E + i]; } else { s = d = i - E; }
    float e = as[s] + ad[d];
    e = e > 0.f ? e : NEG_SLOPE * e;
    atomicMax(&emax_enc[d], enc_f32(e));
}

// ---- 3b. exp(e - max) and denominator --------------------------------------
__global__ void edge_expsum_k(const int* __restrict__ ei, int E, int ET,
                              const float* __restrict__ as,
                              const float* __restrict__ ad,
                              const unsigned* __restrict__ emax_enc,
                              float* __restrict__ ee,
                              float* __restrict__ denom) {
    const int i = blockIdx.x * 256 + threadIdx.x;
    if (i >= ET) return;
    int s, d;
    if (i < E) { s = ei[i]; d = ei[E + i]; } else { s = d = i - E; }
    float e = as[s] + ad[d];
    e = e > 0.f ? e : NEG_SLOPE * e;
    const float v = __expf(e - dec_f32(emax_enc[d]));
    ee[i] = v;
    atomicAdd(&denom[d], v);
}

// ---- 3c. weighted scatter: hagg[dst] += alpha * hl[src] --------------------
// one wave per edge; hl is L2-resident (25.6 MB < 192 MB L2)
__global__ void edge_agg_k(const int* __restrict__ ei, int E, int ET,
                           const float* __restrict__ ee,
                           const float* __restrict__ denom,
                           const float* __restrict__ hl,
                           float* __restrict__ hagg) {
    const int wid  = (blockIdx.x * blockDim.x + threadIdx.x) >> 5;
    const int lane = threadIdx.x & 31;
    if (wid >= ET) return;
    int s, d;
    if (wid < E) { s = ei[wid]; d = ei[E + wid]; } else { s = d = wid - E; }
    const float w = ee[wid] / denom[d];
    const float4 v = ((const float4*)(hl + (size_t)s * C_DIM))[lane];
    float* dst = hagg + (size_t)d * C_DIM + lane * 4;
    atomicAdd(dst + 0, w * v.x);
    atomicAdd(dst + 1, w * v.y);
    atomicAdd(dst + 2, w * v.z);
    atomicAdd(dst + 3, w * v.w);
}

// ---- 4. bias + tanh + batchnorm stats (block-partial reduction) ------------
__global__ void post_tanh_stats_k(const float* __restrict__ hagg,
                                  const float* __restrict__ bias,
                                  float* __restrict__ hact,
                                  float* __restrict__ stats,  // [256]: sum,sumsq
                                  int n) {
    const int c = threadIdx.x & 127;
    const int g = threadIdx.x >> 7;           // 0 or 1
    const int base = blockIdx.x * 32 + g * 16;
    const float bc = bias[c];
    float s = 0.f, ss = 0.f;
    for (int r = 0; r < 16; ++r) {
        const int node = base + r;
        if (node < n) {
            const float t = tanhf(hagg[(size_t)node * C_DIM + c] + bc);
            hact[(size_t)node * C_DIM + c] = t;
            s += t;
            ss += t * t;
        }
    }
    atomicAdd(&stats[c], s);
    atomicAdd(&stats[C_DIM + c], ss);
}

__global__ void bn_apply_k(const float* __restrict__ hact,
                           const float* __restrict__ stats,
                           const float* __restrict__ gamma,
                           const float* __restrict__ beta,
                           float* __restrict__ hout, int n) {
    const int idx = blockIdx.x * 256 + threadIdx.x;
    if (idx >= n * C_DIM) return;
    const int c = idx & 127;
    const float invn = 1.0f / (float)n;
    const float mean = stats[c] * invn;
    const float var  = stats[C_DIM + c] * invn - mean * mean;
    hout[idx] = (hact[idx] - mean) * rsqrtf(var + BN_EPS) * gamma[c] + beta[c];
}

// ---- 5. per-type mean pooling ----------------------------------------------
__global__ void pool_k(const float* __restrict__ h,
                       const int* __restrict__ x_type,
                       const int* __restrict__ batch,
                       float* __restrict__ pool0, float* __restrict__ pool1,
                       float* __restrict__ cnt, int n) {
    const int idx = blockIdx.x * 256 + threadIdx.x;
    const int node = idx >> 7;
    const int c = idx & 127;
    if (node >= n) return;
    const int b = batch[node];
    const int t = x_type[node];
    float* p = (t == 0) ? pool0 : pool1;
    atomicAdd(&p[(size_t)b * C_DIM + c], h[(size_t)node * C_DIM + c]);
    if (c == 0) atomicAdd(&cnt[t * G_DIM + b], 1.0f);
}

__global__ void build_z_k(const float* __restrict__ pool0,
                          const float* __restrict__ pool1,
                          const float* __restrict__ cnt,
                          float* __restrict__ z) {
    const int idx = blockIdx.x * 256 + threadIdx.x;   // G*256
    const int g = idx >> 8;
    const int c = idx & 255;
    if (g >= G_DIM) return;
    float v;
    if (c < C_DIM) {  // x_ab first (type 1)
        v = pool1[(size_t)g * C_DIM + c] / fmaxf(cnt[G_DIM + g], 1.0f);
    } else {          // then x_ag (type 0)
        v = pool0[(size_t)g * C_DIM + (c - C_DIM)] / fmaxf(cnt[g], 1.0f);
    }
    z[(size_t)g * 256 + c] = v;
}

// ---- 6. MLP GEMM (fp32 WMMA) with fused tanh -------------------------------
// grid.x = M/16 row tiles, grid.y * 8 waves = Nc/16 col tiles; M,K,Nc are
// multiples of 16 so no guards anywhere; K-loop fully unrolled.
template <int K, int Nc>
__global__ void mlp_gemm_tanh_wmma(const float* __restrict__ in,
                                   const float* __restrict__ W,
                                   const float* __restrict__ b,
                                   float* __restrict__ out) {
    const int wave = threadIdx.x >> 5;
    const int lane = threadIdx.x & 31;
    const int half = lane >> 4;
    const int r    = lane & 15;
    const int row0 = blockIdx.x * 16;
    const int n0   = (blockIdx.y * 8 + wave) * 16;
    if (n0 >= Nc) return;   // wave-uniform exit
    const float* ap = in + (size_t)(row0 + r) * K + 2 * half;
    const float* bp = W + (size_t)(2 * half) * Nc + n0 + r;
    v8f acc = {};
#pragma unroll
    for (int t = 0; t < K / 4; ++t) {
        v2f a;
        a.x = ap[4 * t];
        a.y = ap[4 * t + 1];
        v2f bb;
        bb.x = bp[(size_t)(4 * t) * Nc];
        bb.y = bp[(size_t)(4 * t + 1) * Nc];
        acc = __builtin_amdgcn_wmma_f32_16x16x4_f32(
            false, a, false, bb, (short)0, acc, false, false);
    }
#pragma unroll
    for (int j = 0; j < 8; ++j) {
        const int m = j + half * 8;
        out[(size_t)(row0 + m) * Nc + n0 + r] = tanhf(acc[j] + b[n0 + r]);
    }
}

// batchnorm over G=512 rows, one thread per channel
__global__ void mlp_bn_k(float* __restrict__ z, const float* __restrict__ gamma,
                         const float* __restrict__ beta, int Nc) {
    const int c = blockIdx.x * blockDim.x + threadIdx.x;
    if (c >= Nc) return;
    float s = 0.f, ss = 0.f;
    for (int g = 0; g < G_DIM; ++g) {
        const float v = z[(size_t)g * Nc + c];
        s += v;
        ss += v * v;
    }
    const float mean = s * (1.0f / G_DIM);
    const float var  = ss * (1.0f / G_DIM) - mean * mean;
    const float inv  = rsqrtf(var + BN_EPS);
    for (int g = 0; g < G_DIM; ++g)
        z[(size_t)g * Nc + c] = (z[(size_t)g * Nc + c] - mean) * inv * gamma[c] + beta[c];
}

// ---- 7. output heads: regression + sigmoid classification ------------------
__global__ void heads_k(const float* __restrict__ z2,   // [512,128]
                        const float* __restrict__ Wr, const float* __restrict__ br,
                        const float* __restrict__ Wc, const float* __restrict__ bc,
                        float* __restrict__ out) {
    const int g = (blockIdx.x * blockDim.x + threadIdx.x) >> 5;
    const int lane = threadIdx.x & 31;
    if (g >= G_DIM) return;
    const float* row = z2 + (size_t)g * C_DIM;
    float sr = 0.f, sc = 0.f;
#pragma unroll
    for (int c = lane; c < C_DIM; c += 32) {
        const float v = row[c];
        sr += v * Wr[c];
        sc += v * Wc[c];
    }
#pragma unroll
    for (int off = 16; off; off >>= 1) {
        sr += __shfl_down(sr, off, 32);
        sc += __shfl_down(sc, off, 32);
    }
    if (lane == 0) {
        out[g] = sr + br[0];
        out[G_DIM + g] = 1.0f / (1.0f + __expf(-(sc + bc[0])));
    }
}

// ---------------------------------------------------------------------------

extern "C" void kernel_launch(void* const* d_in, const int* in_sizes, int n_in,
                              void* d_out, int out_size, void* d_ws, size_t ws_size,
                              hipStream_t stream) {
    const float* x        = (const float*)d_in[0];
    const int*   x_type   = (const int*)  d_in[1];
    const int*   ei       = (const int*)  d_in[2];
    const int*   batch    = (const int*)  d_in[3];
    const float* W_gat    = (const float*)d_in[4];
    const float* att_src  = (const float*)d_in[5];
    const float* att_dst  = (const float*)d_in[6];
    const float* bias_gat = (const float*)d_in[7];
    const float* bn_gamma = (const float*)d_in[8];
    const float* bn_beta  = (const float*)d_in[9];
    const float* W1 = (const float*)d_in[10];
    const float* b1 = (const float*)d_in[11];
    const float* g1 = (const float*)d_in[12];
    const float* be1= (const float*)d_in[13];
    const float* W2 = (const float*)d_in[14];
    const float* b2 = (const float*)d_in[15];
    const float* g2 = (const float*)d_in[16];
    const float* be2= (const float*)d_in[17];
    const float* Wr = (const float*)d_in[18];
    const float* br = (const float*)d_in[19];
    const float* Wc = (const float*)d_in[20];
    const float* bc = (const float*)d_in[21];

    const int N  = in_sizes[0] / C_DIM;
    const int E  = in_sizes[2] / 2;
    const int ET = E + N;
    const int NC = N * C_DIM;

    // ---- workspace carve ----
    float* p = (float*)d_ws;
    float*    hcur  = p; p += NC;
    float*    hact  = p; p += NC;
    float*    hl    = p; p += NC;
    float*    hagg  = p; p += NC;          // >>> zero region start
    float*    denom = p; p += N;
    unsigned* emax  = (unsigned*)p; p += N;
    float*    stats = p; p += 2 * C_DIM;   // <<< zero region end
    const int zero_cnt = NC + 2 * N + 2 * C_DIM;
    float*    alpha_s = p; p += N;
    float*    alpha_d = p; p += N;
    float*    ee      = p; p += ET;
    float*    pool0   = p; p += G_DIM * C_DIM;   // >>> pool zero region
    float*    pool1   = p; p += G_DIM * C_DIM;
    float*    cnt     = p; p += 2 * G_DIM;       // <<<
    const int pool_zero_cnt = 2 * G_DIM * C_DIM + 2 * G_DIM;
    float*    z  = p; p += G_DIM * 256;
    float*    t1 = p; p += G_DIM * 256;
    float*    t2 = p; p += G_DIM * C_DIM;

    const dim3 blk(256);
    const int gemm_grid  = (N + 127) / 128;
    const int node_grid  = (N * 32 + 255) / 256;
    const int edge_grid  = (ET + 255) / 256;
    const int eagg_grid  = (ET * 32 + 255) / 256;
    const int post_grid  = (N + 31) / 32;
    const int elem_grid  = (NC + 255) / 256;

    const float* hin = x;
    for (int layer = 0; layer < 3; ++layer) {
        zero_f32_k<<<(zero_cnt + 255) / 256, blk, 0, stream>>>(hagg, zero_cnt);
        if (layer == 0)
            gat_gemm_wmma<0><<<gemm_grid, blk, 0, stream>>>(
                hin, x, W_gat + (size_t)layer * C_DIM * C_DIM, hl, N);
        else
            gat_gemm_wmma<1><<<gemm_grid, blk, 0, stream>>>(
                hin, x, W_gat + (size_t)layer * C_DIM * C_DIM, hl, N);
        node_alpha_k<<<node_grid, blk, 0, stream>>>(
            hl, att_src + layer * C_DIM, att_dst + layer * C_DIM,
            alpha_s, alpha_d, N);
        edge_max_k<<<edge_grid, blk, 0, stream>>>(ei, E, ET, alpha_s, alpha_d, emax);
        edge_expsum_k<<<edge_grid, blk, 0, stream>>>(ei, E, ET, alpha_s, alpha_d,
                                                     emax, ee, denom);
        edge_agg_k<<<eagg_grid, blk, 0, stream>>>(ei, E, ET, ee, denom, hl, hagg);
        post_tanh_stats_k<<<post_grid, blk, 0, stream>>>(
            hagg, bias_gat + layer * C_DIM, hact, stats, N);
        bn_apply_k<<<elem_grid, blk, 0, stream>>>(
            hact, stats, bn_gamma + layer * C_DIM, bn_beta + layer * C_DIM, hcur, N);
        hin = hcur;
    }

    // pooling + readout
    zero_f32_k<<<(pool_zero_cnt + 255) / 256, blk, 0, stream>>>(pool0, pool_zero_cnt);
    pool_k<<<elem_grid, blk, 0, stream>>>(hcur, x_type, batch, pool0, pool1, cnt, N);
    build_z_k<<<(G_DIM * 256 + 255) / 256, blk, 0, stream>>>(pool0, pool1, cnt, z);

    // MLP: [512,256] @ [256,256] -> tanh -> BN ; then @ [256,128] -> tanh -> BN
    mlp_gemm_tanh_wmma<256, 256><<<dim3(G_DIM / 16, 2), blk, 0, stream>>>(z, W1, b1, t1);
    mlp_bn_k<<<1, 256, 0, stream>>>(t1, g1, be1, 256);
    mlp_gemm_tanh_wmma<256, 128><<<dim3(G_DIM / 16, 1), blk, 0, stream>>>(t1, W2, b2, t2);
    mlp_bn_k<<<1, 128, 0, stream>>>(t2, g2, be2, 128);

    heads_k<<<(G_DIM * 32 + 255) / 256, blk, 0, stream>>>(
        t2, Wr, br, Wc, bc, (float*)d_out);
}